// GeometricLoss_73100343378545
// MI455X (gfx1250) — compile-verified
//
#include <hip/hip_runtime.h>
#include <math.h>

typedef __attribute__((ext_vector_type(2))) float v2f;
typedef __attribute__((ext_vector_type(8))) float v8f;

#define N_PTS 8192
#define C_DIM 16
#define K_NB 5
#define WAVES_PER_BLOCK 8
#define BLOCK_THREADS (WAVES_PER_BLOCK * 32)
#define QUERIES_PER_WAVE 16
#define QUERIES_PER_BLOCK (WAVES_PER_BLOCK * QUERIES_PER_WAVE)   // 128
#define NUM_BLOCKS (N_PTS / QUERIES_PER_BLOCK)                   // 64
#define NUM_CAND_TILES (N_PTS / 16)                              // 512

// Guarded branchless sorted insert of (v, vi) into ascending top-6 (td, ti).
__device__ __forceinline__ void top6_insert(float v, int vi, float td[6], int ti[6]) {
    if (v < td[5]) {
#pragma unroll
        for (int j = 5; j >= 1; --j) {
            bool c  = v < td[j];
            bool cp = v < td[j - 1];
            float nd = cp ? td[j - 1] : v;
            int   ni = cp ? ti[j - 1] : vi;
            td[j] = c ? nd : td[j];
            ti[j] = c ? ni : ti[j];
        }
        bool c0 = v < td[0];
        ti[0] = c0 ? vi : ti[0];
        td[0] = c0 ? v  : td[0];
    }
}

__global__ __launch_bounds__(BLOCK_THREADS)
void GeometricLoss_knn_kernel(const float* __restrict__ outputs,
                              const float* __restrict__ points,
                              float* __restrict__ partials)
{
    // 128 KB: all points resident in LDS, pre-augmented for the K=4 WMMA trick:
    //   saug[i] = { -2x, -2y, x^2+y^2, 1 }
    // A-fragment for candidate m is then exactly 8 contiguous bytes:
    //   lanes 0-15  read saug[m].xy  (K=0,1)
    //   lanes 16-31 read saug[m].zw  (K=2,3)
    __shared__ float4 saug[N_PTS];
    __shared__ float  swave[WAVES_PER_BLOCK];

    const int tid = threadIdx.x;
    const float2* p2 = reinterpret_cast<const float2*>(points);
    for (int i = tid; i < N_PTS; i += BLOCK_THREADS) {
        float2 p = p2[i];
        float4 a;
        a.x = -2.0f * p.x;
        a.y = -2.0f * p.y;
        a.z = p.x * p.x + p.y * p.y;
        a.w = 1.0f;
        saug[i] = a;
    }
    __syncthreads();

    const int wave = tid >> 5;
    const int lane = tid & 31;
    const int col  = lane & 15;        // N index within 16x16 tile
    const bool hi  = lane >= 16;
    const int halfSel = hi ? 1 : 0;

    const int q = (blockIdx.x * WAVES_PER_BLOCK + wave) * QUERIES_PER_WAVE + col;

    // B fragment (queries, 4x16, K split across lane halves):
    //   lanes 0-15 : {x, y}   (K=0,1)
    //   lanes 16-31: {1, sq}  (K=2,3)
    float4 qa = saug[q];
    v2f bfrag;
    bfrag.x = hi ? 1.0f : -0.5f * qa.x;
    bfrag.y = hi ? qa.z : -0.5f * qa.y;

    __builtin_prefetch(outputs + (size_t)q * C_DIM, 0, 3);   // global_prefetch_b8

    float td[6];
    int   ti[6];
#pragma unroll
    for (int j = 0; j < 6; ++j) { td[j] = 3.0e38f; ti[j] = 0; }

    // View the augmented LDS array as float2 halves for direct A-fragment loads.
    const float2* saug2 = reinterpret_cast<const float2*>(saug);

    // Software pipeline: preload tile 0's A-fragment.
    float2 af = saug2[2 * col + halfSel];

    for (int ct = 0; ct < NUM_CAND_TILES; ++ct) {
        v2f afrag;
        afrag.x = af.x;
        afrag.y = af.y;

        // Prefetch next tile's fragment (ds_load hidden behind WMMA + selection).
        const int nct = (ct + 1 < NUM_CAND_TILES) ? (ct + 1) : 0;
        af = saug2[2 * (nct * 16 + col) + halfSel];

        v8f acc = {0.f, 0.f, 0.f, 0.f, 0.f, 0.f, 0.f, 0.f};
        // D[M=candidate][N=query] = squared distance via augmented K=4.
        acc = __builtin_amdgcn_wmma_f32_16x16x4_f32(
            /*neg_a=*/false, afrag, /*neg_b=*/false, bfrag,
            /*c_mod=*/(short)0, acc, /*reuse_a=*/false, /*reuse_b=*/false);

        // Tile-level fast reject: single exec-branch instead of 8.
        float m01 = fminf(acc[0], acc[1]);
        float m23 = fminf(acc[2], acc[3]);
        float m45 = fminf(acc[4], acc[5]);
        float m67 = fminf(acc[6], acc[7]);
        float mall = fminf(fminf(m01, m23), fminf(m45, m67));

        if (mall < td[5]) {
            const int rbase = ct * 16 + (hi ? 8 : 0);
#pragma unroll
            for (int v = 0; v < 8; ++v) {
                top6_insert(acc[v], rbase + v, td, ti);
            }
        }
    }

    // Merge lane L with partner lane L^16 (same query, disjoint candidate halves).
    float pd[6]; int pi[6];
#pragma unroll
    for (int j = 0; j < 6; ++j) {
        pd[j] = __shfl_xor(td[j], 16, 32);
        pi[j] = __shfl_xor(ti[j], 16, 32);
    }
#pragma unroll
    for (int j = 0; j < 6; ++j) top6_insert(pd[j], pi[j], td, ti);

    // Phase 2: slots 1..5 are the 5 nearest non-self neighbors.
    float mysum = 0.0f;
    if (!hi) {
        const float4* oi4 = reinterpret_cast<const float4*>(outputs + (size_t)q * C_DIM);
        float4 a0 = oi4[0], a1 = oi4[1], a2 = oi4[2], a3 = oi4[3];
#pragma unroll
        for (int nb = 1; nb <= K_NB; ++nb) {
            const float4* on4 = reinterpret_cast<const float4*>(outputs + (size_t)ti[nb] * C_DIM);
            float4 b0 = on4[0], b1 = on4[1], b2 = on4[2], b3 = on4[3];
            float s = 0.0f;
            s += (a0.x-b0.x)*(a0.x-b0.x) + (a0.y-b0.y)*(a0.y-b0.y)
               + (a0.z-b0.z)*(a0.z-b0.z) + (a0.w-b0.w)*(a0.w-b0.w);
            s += (a1.x-b1.x)*(a1.x-b1.x) + (a1.y-b1.y)*(a1.y-b1.y)
               + (a1.z-b1.z)*(a1.z-b1.z) + (a1.w-b1.w)*(a1.w-b1.w);
            s += (a2.x-b2.x)*(a2.x-b2.x) + (a2.y-b2.y)*(a2.y-b2.y)
               + (a2.z-b2.z)*(a2.z-b2.z) + (a2.w-b2.w)*(a2.w-b2.w);
            s += (a3.x-b3.x)*(a3.x-b3.x) + (a3.y-b3.y)*(a3.y-b3.y)
               + (a3.z-b3.z)*(a3.z-b3.z) + (a3.w-b3.w)*(a3.w-b3.w);
            mysum += sqrtf(s);
        }
    }

    // Deterministic wave reduction (lanes >= 16 contribute 0).
#pragma unroll
    for (int off = 16; off >= 1; off >>= 1)
        mysum += __shfl_xor(mysum, off, 32);
    if (lane == 0) swave[wave] = mysum;
    __syncthreads();
    if (tid == 0) {
        float s = 0.0f;
#pragma unroll
        for (int w = 0; w < WAVES_PER_BLOCK; ++w) s += swave[w];
        partials[blockIdx.x] = s;
    }
}

__global__ void GeometricLoss_reduce_kernel(const float* __restrict__ partials,
                                            float* __restrict__ out)
{
    __shared__ float sh[NUM_BLOCKS];
    int t = threadIdx.x;
    sh[t] = partials[t];
    __syncthreads();
#pragma unroll
    for (int s = NUM_BLOCKS / 2; s >= 1; s >>= 1) {
        if (t < s) sh[t] += sh[t + s];
        __syncthreads();
    }
    if (t == 0) out[0] = sh[0] * (1.0f / (float)(N_PTS * K_NB));
}

extern "C" void kernel_launch(void* const* d_in, const int* in_sizes, int n_in,
                              void* d_out, int out_size, void* d_ws, size_t ws_size,
                              hipStream_t stream) {
    const float* outputs = (const float*)d_in[0];   // [8192, 16] f32
    const float* points  = (const float*)d_in[1];   // [8192, 2]  f32
    // d_in[2] is k == 5 (compile-time constant per reference)
    float* partials = (float*)d_ws;                 // 64 floats of scratch

    GeometricLoss_knn_kernel<<<NUM_BLOCKS, BLOCK_THREADS, 0, stream>>>(
        outputs, points, partials);
    GeometricLoss_reduce_kernel<<<1, NUM_BLOCKS, 0, stream>>>(
        partials, (float*)d_out);
}